// ReconstructionLoss_39848706573634
// MI455X (gfx1250) — compile-verified
//
#include <hip/hip_runtime.h>
#include <math.h>

#define NPTS 16384
#define TILE 16
#define NT   (NPTS / TILE)      // 1024 tiles per dimension
#define ROWS_PER_WAVE 32        // two 16-row A tiles per wave
#define ROWS_PER_WG 256         // 8 waves * 32 rows
#define WAVES_PER_WG 8
#define POSINF_BITS 0x7F800000

typedef __attribute__((ext_vector_type(2))) float v2f;
typedef __attribute__((ext_vector_type(8))) float v8f;

static __device__ __forceinline__ int imin(int a, int b) { return a < b ? a : b; }
static __device__ __forceinline__ int imax(int a, int b) { return a > b ? a : b; }

// ---------------------------------------------------------------------------
// Kernel 1: pack xyz + squared norms into float4 arrays, init min buffers.
//   pred4[i] = (-2x, -2y, -2z, |p|^2)   (A-operand form, p2 rides in K=3)
//   gt4[i]   = ( x,   y,   z,  |g|^2)   (B-operand form)
// ---------------------------------------------------------------------------
__global__ __launch_bounds__(256)
void rl_pack_kernel(const float* __restrict__ pred,
                    const float* __restrict__ gt,
                    float4* __restrict__ pred4,
                    float4* __restrict__ gt4,
                    unsigned int* __restrict__ rowminsq_u,
                    unsigned int* __restrict__ colminsq_u) {
  int i = blockIdx.x * blockDim.x + threadIdx.x;
  if (i >= NPTS) return;
  float px = pred[i * 6 + 0], py = pred[i * 6 + 1], pz = pred[i * 6 + 2];
  float gx = gt[i * 6 + 0],   gy = gt[i * 6 + 1],   gz = gt[i * 6 + 2];
  pred4[i] = make_float4(-2.0f * px, -2.0f * py, -2.0f * pz,
                         px * px + py * py + pz * pz);
  gt4[i] = make_float4(gx, gy, gz, gx * gx + gy * gy + gz * gz);
  rowminsq_u[i] = (unsigned int)POSINF_BITS;
  colminsq_u[i] = (unsigned int)POSINF_BITS;
}

// ---------------------------------------------------------------------------
// Kernel 2: tiled squared-distance matrix via V_WMMA_F32_16X16X4_F32.
//   A[M][0..2] = -2*pred_xyz,  A[M][3] = |p|^2
//   B[0..2][N] =  gt_xyz,      B[3][N] = 1.0
//   C[m][n]    = |g_n|^2 (splat of gb.w per lane)
//   => D = |p|^2 + |g|^2 - 2 p.g = squared-distance tile, one WMMA each.
// Each wave owns TWO 16-row A tiles; the B operand and C splat are shared by
// both WMMAs per gt tile, their column mins merge into one ds_min_u32.
// All min reductions run in the SIGNED-INT bit-pattern domain (v_min_i32):
// identical ordering to float for non-negative values, and any negative
// candidate (fp rounding of a near-zero distance) drives the result
// negative, which the single final imax(.,0) clamps to 0 — exactly matching
// the reference's max(sq,0)-before-min. This eliminates all float-min
// canonicalization VALU ops. sqrt is hoisted out of the O(N^2) loop
// (min(sqrt(x)) == sqrt(min(x))). gt loads are software-pipelined.
// ---------------------------------------------------------------------------
__global__ __launch_bounds__(256)
void rl_chamfer_tiles_kernel(const float4* __restrict__ pred4,
                             const float4* __restrict__ gt4,
                             unsigned int* __restrict__ rowminsq_u,
                             unsigned int* __restrict__ colminsq_u) {
  __shared__ unsigned int lds_cmin[NPTS];  // 64 KB of the 320 KB WGP LDS

  const int tid  = threadIdx.x;
  const int lane = tid & 31;
  const int wave = tid >> 5;
  const int half = (lane >= 16) ? 1 : 0;  // upper half-wave: K={2,3}, M=8..15
  const int l15  = lane & 15;

  for (int c = tid; c < NPTS; c += 256) lds_cmin[c] = (unsigned int)POSINF_BITS;
  __syncthreads();

  const int i0 = blockIdx.x * ROWS_PER_WG + wave * ROWS_PER_WAVE;
  const int i1 = i0 + TILE;

  // Two A operands (16x4 f32 each) for rows M = i0/i1 + (lane&15):
  //   half0: (K0,K1) = (-2x, -2y);  half1: (K2,K3) = (-2z, p2)
  float4 pa0 = pred4[i0 + l15];
  float4 pa1 = pred4[i1 + l15];
  v2f a0, a1;
  a0.x = half ? pa0.z : pa0.x;
  a0.y = half ? pa0.w : pa0.y;
  a1.x = half ? pa1.z : pa1.x;
  a1.y = half ? pa1.w : pa1.y;

  int rmin0[8], rmin1[8];
#pragma unroll
  for (int r = 0; r < 8; ++r) {
    rmin0[r] = POSINF_BITS;
    rmin1[r] = POSINF_BITS;
  }

  const int tstart = wave * (NT / WAVES_PER_WG);  // stagger waves
  // Software pipeline: prefetch tile t, compute while loading t+1.
  float4 gb = gt4[(tstart & (NT - 1)) * TILE + l15];

  for (int t = 0; t < NT; ++t) {
    const int j0  = ((t + tstart) & (NT - 1)) * TILE;
    const int j0n = ((t + 1 + tstart) & (NT - 1)) * TILE;
    float4 gbn = gt4[j0n + l15];  // next tile (wraps harmlessly on last iter)

    // B operand (4x16 f32) for col N = j0 + (lane&15):
    //   half0: (K0,K1) = (x, y);  half1: (K2,K3) = (z, 1.0)
    v2f b;
    b.x = half ? gb.z : gb.x;
    b.y = half ? 1.0f : gb.y;

    v8f c;
#pragma unroll
    for (int r = 0; r < 8; ++r) c[r] = gb.w;  // + |g|^2, shared by both WMMAs

    v8f d0 = __builtin_amdgcn_wmma_f32_16x16x4_f32(
        false, a0, false, b, (short)0, c, false, false);
    v8f d1 = __builtin_amdgcn_wmma_f32_16x16x4_f32(
        false, a1, false, b, (short)0, c, false, false);

    // Row mins (running, per lane) + column min across both 16-row tiles,
    // all in the signed-int bit-pattern domain (no canonicalization ops).
    int cm = POSINF_BITS;
#pragma unroll
    for (int r = 0; r < 8; ++r) {
      const int e0 = __float_as_int(d0[r]);
      const int e1 = __float_as_int(d1[r]);
      rmin0[r] = imin(rmin0[r], e0);
      rmin1[r] = imin(rmin1[r], e1);
      cm = imin(cm, imin(e0, e1));
    }
    cm = imax(cm, 0);  // clamp-to-zero == reference's max(sq,0) before min
    // Both half-waves hit the same 16 addresses; ds_min_u32 merges them.
    atomicMin(&lds_cmin[j0 + l15], (unsigned int)cm);

    gb = gbn;
  }

  // Reduce row mins across the 16 lanes of each half-wave (int domain).
#pragma unroll
  for (int r = 0; r < 8; ++r) {
    int v0 = rmin0[r];
    int v1 = rmin1[r];
#pragma unroll
    for (int m = 1; m < 16; m <<= 1) {
      v0 = imin(v0, __shfl_xor(v0, m, 32));
      v1 = imin(v1, __shfl_xor(v1, m, 32));
    }
    if (l15 == 0) {
      rowminsq_u[i0 + r + half * 8] = (unsigned int)imax(v0, 0);
      rowminsq_u[i1 + r + half * 8] = (unsigned int)imax(v1, 0);
    }
  }

  // Flush workgroup-local column mins to global (one atomic per column).
  __syncthreads();
  for (int c = tid; c < NPTS; c += 256)
    atomicMin(&colminsq_u[c], lds_cmin[c]);
}

// ---------------------------------------------------------------------------
// Kernel 3: sqrt + means of row/col mins, L1 color loss, weighted combine.
// ---------------------------------------------------------------------------
__global__ __launch_bounds__(256)
void rl_finalize_kernel(const unsigned int* __restrict__ rowminsq_u,
                        const unsigned int* __restrict__ colminsq_u,
                        const float* __restrict__ pred,
                        const float* __restrict__ gt,
                        float* __restrict__ out) {
  __shared__ float s0[256], s1[256], s2[256];
  const int tid = threadIdx.x;
  float rs = 0.0f, cs = 0.0f, col = 0.0f;
  for (int i = tid; i < NPTS; i += 256) {
    rs += sqrtf(__uint_as_float(rowminsq_u[i]));
    cs += sqrtf(__uint_as_float(colminsq_u[i]));
    col += fabsf(pred[i * 6 + 3] - gt[i * 6 + 3]) +
           fabsf(pred[i * 6 + 4] - gt[i * 6 + 4]) +
           fabsf(pred[i * 6 + 5] - gt[i * 6 + 5]);
  }
  s0[tid] = rs; s1[tid] = cs; s2[tid] = col;
  __syncthreads();
  for (int off = 128; off > 0; off >>= 1) {
    if (tid < off) {
      s0[tid] += s0[tid + off];
      s1[tid] += s1[tid + off];
      s2[tid] += s2[tid + off];
    }
    __syncthreads();
  }
  if (tid == 0) {
    const float chamfer = s0[0] / (float)NPTS + s1[0] / (float)NPTS;
    const float color   = s2[0] / (float)(NPTS * 3);
    out[0] = 1.0f * chamfer + 0.1f * color;
  }
}

// ---------------------------------------------------------------------------
extern "C" void kernel_launch(void* const* d_in, const int* in_sizes, int n_in,
                              void* d_out, int out_size, void* d_ws, size_t ws_size,
                              hipStream_t stream) {
  const float* pred = (const float*)d_in[0];  // [16384, 6]
  const float* gt   = (const float*)d_in[1];  // [16384, 6]
  float* out = (float*)d_out;

  // Workspace carve-up: pred4(4N) | gt4(4N) | rowminsq(N) | colminsq(N)
  float* ws = (float*)d_ws;
  float4* pred4 = (float4*)ws;
  float4* gt4   = (float4*)(ws + 4 * NPTS);
  unsigned int* rowminsq_u = (unsigned int*)(ws + 8 * NPTS);
  unsigned int* colminsq_u = (unsigned int*)(ws + 9 * NPTS);

  rl_pack_kernel<<<NPTS / 256, 256, 0, stream>>>(pred, gt, pred4, gt4,
                                                 rowminsq_u, colminsq_u);
  rl_chamfer_tiles_kernel<<<NPTS / ROWS_PER_WG, 256, 0, stream>>>(
      pred4, gt4, rowminsq_u, colminsq_u);
  rl_finalize_kernel<<<1, 256, 0, stream>>>(rowminsq_u, colminsq_u, pred, gt, out);
}